// GCMCLayer_51049981280528
// MI455X (gfx1250) — compile-verified
//
#include <hip/hip_runtime.h>

typedef __attribute__((ext_vector_type(2))) float v2f;
typedef __attribute__((ext_vector_type(8))) float v8f;

// ---------------------------------------------------------------------------
// Zero the f32 accumulators in workspace.
// ---------------------------------------------------------------------------
__global__ void gcmc_zero_kernel(float* __restrict__ p, long n) {
  long i = (long)blockIdx.x * blockDim.x + threadIdx.x;
  long stride = (long)gridDim.x * blockDim.x;
  for (; i < n; i += stride) p[i] = 0.0f;
}

// ---------------------------------------------------------------------------
// Gather-scale-scatter: for each (rating r, edge e), thread t in [0,64):
//   acc[dst*64 + t] += W[(r*Nsrc + src)*64 + t] * cj[src]
// 64-thread groups -> 256B coalesced W-row reads; no-return f32 atomics to L2.
// D is fixed at 64 by the reference (shifts <<6).
// ---------------------------------------------------------------------------
__global__ void gcmc_scatter_kernel(const int* __restrict__ e_src,
                                    const int* __restrict__ e_dst,
                                    const float* __restrict__ W,
                                    const float* __restrict__ cj,
                                    float* __restrict__ acc,
                                    int E, int Nsrc, long total) {
  long gid = (long)blockIdx.x * blockDim.x + threadIdx.x;
  if (gid >= total) return;
  int t = (int)(gid & 63);
  long edge = gid >> 6;            // flat index into [R, E]
  int src = e_src[edge];
  int dst = e_dst[edge];
  long r = edge / E;
  float c = cj[src];
  float w = W[((r * Nsrc + src) << 6) + t];
  atomicAdd(&acc[((long)dst << 6) + t], w * c);
}

// ---------------------------------------------------------------------------
// Per-node FC with f32 WMMA:  out[N,64] = leaky(acc[N,64] * ci[N]) @ fcw + fcb
// One wave32 per 16-row node tile; computes 16x64 output via 4 N-tiles,
// K looped as 16 chunks of 4 using V_WMMA_F32_16X16X4_F32.
//
// A(16x4 f32) layout: lane(0-15): M=lane, {v0,v1}={K0,K1}; lane(16-31): {K2,K3}
// B(4x16 f32) layout: v0 = rows K0(lanes 0-15)/K2(16-31), v1 = rows K1/K3
// C/D(16x16)  layout: VGPR v: lanes 0-15 -> (M=v, N=lane); 16-31 -> (M=v+8)
// ---------------------------------------------------------------------------
__global__ void gcmc_fc_wmma_kernel(const float* __restrict__ acc,  // [N,64]
                                    const float* __restrict__ ci,   // [N]
                                    const float* __restrict__ fcw,  // [64,64]
                                    const float* __restrict__ fcb,  // [64]
                                    float* __restrict__ out,        // [N,64]
                                    int N) {
  const int lane = threadIdx.x & 31;
  const int wave = threadIdx.x >> 5;
  const int tile = blockIdx.x * (blockDim.x >> 5) + wave;
  const int row0 = tile * 16;
  if (row0 >= N) return;                 // uniform per wave: EXEC stays full

  const int m  = lane & 15;
  const int kh = lane >> 4;              // 0: K even pair, 1: K odd pair
  int row = row0 + m;
  int rowc = row < N ? row : N - 1;      // clamp (N%16==0 in practice)
  const float cis = ci[rowc];

  // Preload full per-lane A strip: 16 kb-chunks x 2 f32, with ci + LeakyReLU fused.
  v2f a[16];
  const float* arow = acc + ((long)rowc << 6);
  #pragma unroll
  for (int kb = 0; kb < 16; ++kb) {
    float x0 = arow[kb * 4 + kh * 2 + 0] * cis;
    float x1 = arow[kb * 4 + kh * 2 + 1] * cis;
    a[kb].x = x0 >= 0.0f ? x0 : 0.1f * x0;
    a[kb].y = x1 >= 0.0f ? x1 : 0.1f * x1;
  }

  #pragma unroll
  for (int nt = 0; nt < 4; ++nt) {
    const int ncol = nt * 16 + (lane & 15);
    // Init C with the bias (depends on N only -> identical across M rows).
    v8f c;
    const float b0 = fcb[ncol];
    #pragma unroll
    for (int v = 0; v < 8; ++v) c[v] = b0;

    #pragma unroll
    for (int kb = 0; kb < 16; ++kb) {
      const int k = kb * 4 + kh * 2;
      v2f b;
      b.x = fcw[(k + 0) * 64 + ncol];
      b.y = fcw[(k + 1) * 64 + ncol];
      c = __builtin_amdgcn_wmma_f32_16x16x4_f32(
          /*neg_a=*/false, a[kb], /*neg_b=*/false, b,
          /*c_mod=*/(short)0, c, /*reuse_a=*/false, /*reuse_b=*/false);
    }

    #pragma unroll
    for (int v = 0; v < 8; ++v) {
      const int orow = row0 + v + kh * 8;
      if (orow < N) out[(long)orow * 64 + ncol] = c[v];
    }
  }
}

// ---------------------------------------------------------------------------
// Launch
// ---------------------------------------------------------------------------
extern "C" void kernel_launch(void* const* d_in, const int* in_sizes, int n_in,
                              void* d_out, int out_size, void* d_ws, size_t ws_size,
                              hipStream_t stream) {
  const int*   edge_u = (const int*)d_in[0];
  const int*   edge_i = (const int*)d_in[1];
  const float* cj_u   = (const float*)d_in[2];
  const float* ci_u   = (const float*)d_in[3];
  const float* cj_i   = (const float*)d_in[4];
  const float* ci_i   = (const float*)d_in[5];
  const float* W_user = (const float*)d_in[6];
  const float* W_item = (const float*)d_in[7];
  const float* ufc_w  = (const float*)d_in[8];
  const float* ufc_b  = (const float*)d_in[9];
  const float* ifc_w  = (const float*)d_in[10];
  const float* ifc_b  = (const float*)d_in[11];
  float* out = (float*)d_out;

  const int  NU  = in_sizes[2];
  const int  NI  = in_sizes[4];
  const int  OUT = in_sizes[9];
  const int  D   = in_sizes[8] / OUT;                         // 64
  const long RE  = (long)in_sizes[0];                         // R*E
  const int  R   = (int)((long)in_sizes[6] / ((long)NU * D)); // 5
  const int  E   = (int)(RE / R);

  float* acc_u = (float*)d_ws;                 // [NU, 64] ufeat accumulator
  float* acc_i = acc_u + (long)NU * D;         // [NI, 64] ifeat accumulator

  // 1) zero accumulators
  const long accN = (long)(NU + NI) * D;
  gcmc_zero_kernel<<<1024, 256, 0, stream>>>(acc_u, accN);

  // 2) forward (user -> item) and reverse (item -> user) scatter passes
  const long total = RE * 64;
  const unsigned sblocks = (unsigned)((total + 255) / 256);
  gcmc_scatter_kernel<<<sblocks, 256, 0, stream>>>(edge_u, edge_i, W_user,
                                                   cj_u, acc_i, E, NU, total);
  gcmc_scatter_kernel<<<sblocks, 256, 0, stream>>>(edge_i, edge_u, W_item,
                                                   cj_i, acc_u, E, NI, total);

  // 3) per-type FC with f32 WMMA (8 waves/block -> 8 row-tiles per block)
  const int tilesU = (NU + 15) / 16;
  const int tilesI = (NI + 15) / 16;
  gcmc_fc_wmma_kernel<<<(tilesU + 7) / 8, 256, 0, stream>>>(
      acc_u, ci_u, ufc_w, ufc_b, out, NU);
  gcmc_fc_wmma_kernel<<<(tilesI + 7) / 8, 256, 0, stream>>>(
      acc_i, ci_i, ifc_w, ifc_b, out + (long)NU * OUT, NI);
}